// BandUpsample_12249246728376
// MI455X (gfx1250) — compile-verified
//
#include <hip/hip_runtime.h>
#include <cstddef>

#if defined(__HIP_DEVICE_COMPILE__)
#if !__has_builtin(__builtin_amdgcn_wmma_f32_16x16x4_f32)
#error "gfx1250 f32 WMMA builtin not available on device"
#endif
#endif

typedef float v2f __attribute__((ext_vector_type(2)));
typedef float v8f __attribute__((ext_vector_type(8)));

#define CHN 64
#define BATCH 64
#define T_IN 32
#define N_OSC 32
#define BAND 16384
#define NFRAMES 4096
#define LOWEST_FREQ 0.0018140589569160998f
#define PI_F 3.14159265358979323846f

// Padded LDS row stride: 68 floats -> bank index (4*row + col) % 64, so the
// 32 lanes' 8-byte fragment loads tile all 64 LDS banks conflict-free.
#define WPAD 68

// ---------------------------------------------------------------------------
// Generic 3-tap conv1d, Cin=Cout=64, optional fused nearest-2x upsample of the
// input, optional leaky-relu.  One workgroup = (batch b, 32-sample out chunk),
// 8 waves, each wave owns one 16x16 output tile computed with
// v_wmma_f32_16x16x4_f32 (3 taps x 16 K-steps = 48 WMMA per tile).
// WMMA f32 lane layouts (ISA 7.12.2):
//   A 16x4 : VGPR j, lane L -> M = L%16,  K = j + 2*(L>=16)
//   B 4x16 : VGPR j, lane L -> N = L%16,  K = j + 2*(L>=16)   (mirror of A)
//   C/D    : VGPR v, lane L -> M = v + 8*(L>=16), N = L%16
// A fragment = one b64 LDS load from Wlds[tap][cout][kk..kk+1]
// B fragment = one b64 LDS load from Ilds[t][kk..kk+1]  (input stored t-major)
// ---------------------------------------------------------------------------
__global__ __launch_bounds__(256)
void conv3_wmma_kernel(const float* __restrict__ in, float* __restrict__ out,
                       const float* __restrict__ w, const float* __restrict__ bias,
                       int T_out, int in_sb, int in_sc, int in_st,
                       int up, int leaky, int nchunks)
{
    __shared__ float Wlds[3][CHN][WPAD];  // [tap][cout][cin]   52.2 KB
    __shared__ float Ilds[34][WPAD];      // [t][cin] padded     9.2 KB

    const int b  = blockIdx.x / nchunks;
    const int c0 = (blockIdx.x % nchunks) * 32;
    const int tid = threadIdx.x;

    // stage weights: flat w index = o*192 + ci*3 + k
    for (int i = tid; i < CHN * CHN * 3; i += 256) {
        int k = i % 3, r = i / 3, ci = r % CHN, o = r / CHN;
        Wlds[k][o][ci] = w[i];
    }
    // stage input samples t in [c0-1, c0+33); keep t fastest-varying for
    // coalesced global reads, write transposed into Ilds[t][c].
    const float* inb = in + (size_t)b * in_sb;
    for (int i = tid; i < CHN * 34; i += 256) {
        int row = i / 34, col = i % 34;      // row = cin, col = t - (c0-1)
        int p = c0 - 1 + col;
        float v = 0.0f;
        if (p >= 0 && p < T_out) {
            int ts = up ? (p >> 1) : p;
            v = inb[row * in_sc + ts * in_st];
        }
        Ilds[col][row] = v;
    }
    __syncthreads();

    const int wid  = tid >> 5;
    const int lane = tid & 31;
    const int mt = wid >> 1;          // 0..3  -> cout tile
    const int nt = wid & 1;           // 0..1  -> time tile
    const int hi = lane >> 4;         // half-wave select
    const int lm = lane & 15;

    const int cout_a = mt * 16 + lm;  // A-row for this lane
    v8f acc = {0.f, 0.f, 0.f, 0.f, 0.f, 0.f, 0.f, 0.f};

    for (int tap = 0; tap < 3; ++tap) {
        const int tb = nt * 16 + lm + tap;     // row in Ilds (t - (c0-1))
        #pragma unroll
        for (int kb = 0; kb < 16; ++kb) {
            const int kk = kb * 4 + 2 * hi;    // even -> 8B aligned
            v2f a  = *(const v2f*)&Wlds[tap][cout_a][kk];
            v2f bv = *(const v2f*)&Ilds[tb][kk];
            acc = __builtin_amdgcn_wmma_f32_16x16x4_f32(
                      false, a, false, bv, (short)0, acc, false, false);
        }
    }

    float* outb = out + (size_t)b * CHN * T_out;
    const int t = c0 + nt * 16 + lm;
    if (t < T_out) {
        #pragma unroll
        for (int v = 0; v < 8; ++v) {
            int co = mt * 16 + v + 8 * hi;
            float val = acc[v] + bias[co];
            if (leaky) val = val > 0.0f ? val : 0.2f * val;
            outb[co * T_out + t] = val;
        }
    }
}

// ---------------------------------------------------------------------------
// Oscillator params: amp = |z^T aw + ab|, freq = low + sigmoid(z^T fw + fb)*(1-low)
// one thread per (b, osc, t)
// ---------------------------------------------------------------------------
__global__ __launch_bounds__(256)
void oscparam_kernel(const float* __restrict__ z,
                     const float* __restrict__ aw, const float* __restrict__ ab,
                     const float* __restrict__ fw, const float* __restrict__ fb,
                     float* __restrict__ F, float* __restrict__ AMP)
{
    int id = blockIdx.x * 256 + threadIdx.x;            // (b*32+o)*32 + t
    if (id >= BATCH * N_OSC * T_IN) return;
    int t = id & 31, o = (id >> 5) & 31, b = id >> 10;
    const float* zb = z + (size_t)b * CHN * T_IN + t;
    float da = 0.f, df = 0.f;
    for (int c = 0; c < CHN; ++c) {
        float zv = zb[c * T_IN];
        da += zv * aw[o * CHN + c];
        df += zv * fw[o * CHN + c];
    }
    float amp = fabsf(da + ab[o]);
    float sg  = 1.0f / (1.0f + __expf(-(df + fb[o])));
    F[id]   = LOWEST_FREQ + sg * (1.0f - LOWEST_FREQ);
    AMP[id] = amp;
}

// ---------------------------------------------------------------------------
// Closed-form phase prefix: cumulative freq-sum at the 33 interp-segment
// boundaries.  One thread per (b, osc).
// ---------------------------------------------------------------------------
__global__ __launch_bounds__(256)
void phasebase_kernel(const float* __restrict__ F, float* __restrict__ BASE)
{
    int id = blockIdx.x * 256 + threadIdx.x;            // b*32 + osc
    if (id >= BATCH * N_OSC) return;
    const float* f = F + id * 32;
    float* bs = BASE + id * 33;
    bs[0] = 0.0f;
    float acc = 256.0f * f[0];                           // seg 0: 256 samples @ F0
    bs[1] = acc;
    for (int s = 1; s <= 31; ++s) {                      // 512-sample linear segs
        acc += 256.0f * (f[s - 1] + f[s]);
        bs[s + 1] = acc;
    }
}

// ---------------------------------------------------------------------------
// 1x1 conv z -> noise-model seed  (b, cout, t)
// ---------------------------------------------------------------------------
__global__ __launch_bounds__(256)
void nzinit_kernel(const float* __restrict__ z, const float* __restrict__ w,
                   const float* __restrict__ bias, float* __restrict__ y)
{
    int id = blockIdx.x * 256 + threadIdx.x;            // (b*64+co)*32 + t
    if (id >= BATCH * CHN * T_IN) return;
    int t = id & 31, co = (id >> 5) & 63, b = id >> 11;
    const float* zb = z + (size_t)b * CHN * T_IN + t;
    float d = 0.f;
    for (int c = 0; c < CHN; ++c) d += zb[c * T_IN] * w[co * CHN + c];
    y[id] = d + bias[co];
}

// ---------------------------------------------------------------------------
// 1x1 conv y_final -> 3 squared rfft coefficients per frame
// ---------------------------------------------------------------------------
__global__ __launch_bounds__(256)
void coeff_kernel(const float* __restrict__ y, const float* __restrict__ w,
                  const float* __restrict__ bias, float* __restrict__ coeffs)
{
    int id = blockIdx.x * 256 + threadIdx.x;            // b*4096 + frame
    if (id >= BATCH * NFRAMES) return;
    int frame = id & (NFRAMES - 1), b = id >> 12;
    const float* yb = y + (size_t)b * CHN * NFRAMES + frame;
    float d0 = 0.f, d1 = 0.f, d2 = 0.f;
    for (int c = 0; c < CHN; ++c) {
        float v = yb[c * NFRAMES];
        d0 += v * w[0 * CHN + c];
        d1 += v * w[1 * CHN + c];
        d2 += v * w[2 * CHN + c];
    }
    d0 += bias[0]; d1 += bias[1]; d2 += bias[2];
    coeffs[id * 3 + 0] = d0 * d0;
    coeffs[id * 3 + 1] = d1 * d1;
    coeffs[id * 3 + 2] = d2 * d2;
}

// ---------------------------------------------------------------------------
// Final kernel: harmonic bank (closed-form phase, 32 sins/sample) + filtered
// noise (4-point rfft/irfft folded into FMAs).  One thread per output sample;
// per-batch osc tables staged in LDS.  256 contiguous samples per WG always
// fall inside one interpolation segment (segments are 512 long, 256-aligned).
// ---------------------------------------------------------------------------
__global__ __launch_bounds__(256)
void harm_noise_kernel(const float* __restrict__ F, const float* __restrict__ AMP,
                       const float* __restrict__ BASE,
                       const float* __restrict__ coeffs,
                       const float* __restrict__ noise, float* __restrict__ out)
{
    __shared__ float sF[N_OSC * 32];
    __shared__ float sA[N_OSC * 32];
    __shared__ float sB[N_OSC * 33];

    const int b     = blockIdx.x >> 6;      // 64 chunks of 256 samples per batch
    const int chunk = blockIdx.x & 63;
    const int tid   = threadIdx.x;

    for (int i = tid; i < N_OSC * 32; i += 256) {
        sF[i] = F[b * N_OSC * 32 + i];
        sA[i] = AMP[b * N_OSC * 32 + i];
    }
    for (int i = tid; i < N_OSC * 33; i += 256) sB[i] = BASE[b * N_OSC * 33 + i];
    __syncthreads();

    const int n = chunk * 256 + tid;

    // linear-interp coords (align_corners=False), clipped
    float cc = (n + 0.5f) * (1.0f / 512.0f) - 0.5f;
    float cl = fminf(fmaxf(cc, 0.0f), 31.0f);
    int lo = (int)cl; if (lo > 31) lo = 31;
    int hi = lo < 31 ? lo + 1 : 31;
    float frac = cl - (float)lo;

    // phase segment: 0 = [0,256) const, 1..31 = 512-sample linear, 32 = tail const
    int s, segstart;
    if (n < 256)            { s = 0;  segstart = 0; }
    else if (n >= 16128)    { s = 32; segstart = 16128; }
    else                    { s = 1 + (n - 256) / 512; segstart = 256 + 512 * (s - 1); }
    const float ip1  = (float)(n - segstart + 1);
    const float quad = (s >= 1 && s <= 31) ? (ip1 * ip1 * (1.0f / 1024.0f)) : 0.0f;

    float harm = 0.0f;
    #pragma unroll 4
    for (int o = 0; o < N_OSC; ++o) {
        float fl = sF[o * 32 + lo];
        float fh = sF[o * 32 + hi];
        float amp = sA[o * 32 + lo] * (1.0f - frac) + sA[o * 32 + hi] * frac;
        float P = sB[o * 33 + s] + ip1 * fl + (fh - fl) * quad;
        harm += amp * __sinf(PI_F * P);
    }

    // noise: frame = n>>2, 4-pt rfft * squared coeffs * irfft (norm='ortho')
    const int frame = n >> 2, j = n & 3;
    const float* yy = noise + ((size_t)b * NFRAMES + frame) * 4;
    const float* cf = coeffs + ((size_t)b * NFRAMES + frame) * 3;
    float y0 = yy[0], y1 = yy[1], y2 = yy[2], y3 = yy[3];
    float Z0 = cf[0] * 0.5f * (y0 + y1 + y2 + y3);
    float ar = cf[1] * 0.5f * (y0 - y2);
    float ai = cf[1] * 0.5f * (y3 - y1);
    float Z2 = cf[2] * 0.5f * (y0 - y1 + y2 - y3);
    float nz;
    if (j == 0)      nz = 0.5f * (Z0 + 2.0f * ar + Z2);
    else if (j == 1) nz = 0.5f * (Z0 - 2.0f * ai - Z2);
    else if (j == 2) nz = 0.5f * (Z0 - 2.0f * ar + Z2);
    else             nz = 0.5f * (Z0 + 2.0f * ai - Z2);

    out[(size_t)b * BAND + n] = harm + nz;
}

// ---------------------------------------------------------------------------
extern "C" void kernel_launch(void* const* d_in, const int* in_sizes, int n_in,
                              void* d_out, int out_size, void* d_ws, size_t ws_size,
                              hipStream_t stream)
{
    (void)in_sizes; (void)n_in; (void)out_size; (void)ws_size;
    const float* x        = (const float*)d_in[0];
    const float* noise    = (const float*)d_in[1];
    const float* net_w    = (const float*)d_in[2];
    const float* net_b    = (const float*)d_in[3];
    const float* final_w  = (const float*)d_in[4];
    const float* final_b  = (const float*)d_in[5];
    const float* amp_w    = (const float*)d_in[6];
    const float* amp_b    = (const float*)d_in[7];
    const float* freq_w   = (const float*)d_in[8];
    const float* freq_b   = (const float*)d_in[9];
    const float* nzi_w    = (const float*)d_in[10];
    const float* nzi_b    = (const float*)d_in[11];
    const float* nzu_w    = (const float*)d_in[12];
    const float* nzu_b    = (const float*)d_in[13];
    const float* nzf_w    = (const float*)d_in[14];
    const float* nzf_b    = (const float*)d_in[15];
    float* outp = (float*)d_out;

    // workspace carve (floats)
    float* p   = (float*)d_ws;
    float* zA  = p; p += BATCH * CHN * T_IN;
    float* zB  = p; p += BATCH * CHN * T_IN;
    float* Fb  = p; p += BATCH * N_OSC * 32;
    float* Ab  = p; p += BATCH * N_OSC * 32;
    float* Bb  = p; p += BATCH * N_OSC * 33;
    float* cof = p; p += BATCH * NFRAMES * 3;
    float* nA  = p; p += (size_t)BATCH * CHN * NFRAMES;
    float* nB  = p; p += (size_t)BATCH * CHN * NFRAMES;

    // ---- conv stack (T=32, 1 chunk per batch) ----
    // layer 0 reads x in (B,T,CH) layout -> sc=1, st=64
    conv3_wmma_kernel<<<BATCH, 256, 0, stream>>>(x,  zA, net_w + 0 * 12288, net_b + 0 * 64,
                                                 32, T_IN * CHN, 1, CHN, 0, 1, 1);
    conv3_wmma_kernel<<<BATCH, 256, 0, stream>>>(zA, zB, net_w + 1 * 12288, net_b + 1 * 64,
                                                 32, CHN * T_IN, T_IN, 1, 0, 1, 1);
    conv3_wmma_kernel<<<BATCH, 256, 0, stream>>>(zB, zA, net_w + 2 * 12288, net_b + 2 * 64,
                                                 32, CHN * T_IN, T_IN, 1, 0, 1, 1);
    conv3_wmma_kernel<<<BATCH, 256, 0, stream>>>(zA, zB, net_w + 3 * 12288, net_b + 3 * 64,
                                                 32, CHN * T_IN, T_IN, 1, 0, 1, 1);
    conv3_wmma_kernel<<<BATCH, 256, 0, stream>>>(zB, zA, final_w, final_b,
                                                 32, CHN * T_IN, T_IN, 1, 0, 0, 1);
    // z_final = zA

    // ---- oscillator bank precompute ----
    oscparam_kernel<<<(BATCH * N_OSC * T_IN + 255) / 256, 256, 0, stream>>>(
        zA, amp_w, amp_b, freq_w, freq_b, Fb, Ab);
    phasebase_kernel<<<(BATCH * N_OSC + 255) / 256, 256, 0, stream>>>(Fb, Bb);

    // ---- noise model ----
    nzinit_kernel<<<(BATCH * CHN * T_IN + 255) / 256, 256, 0, stream>>>(zA, nzi_w, nzi_b, nA);
    float* nin = nA; float* nout = nB;
    for (int i = 0; i < 7; ++i) {
        int T  = 64 << i;          // output length of this level
        int Ti = T >> 1;           // input length
        int nch = T / 32;
        conv3_wmma_kernel<<<BATCH * nch, 256, 0, stream>>>(
            nin, nout, nzu_w + (size_t)i * 12288, nzu_b + i * 64,
            T, CHN * Ti, Ti, 1, 1, 1, nch);
        float* t = nin; nin = nout; nout = t;
    }
    // final y lives in `nin` after the last swap (= nB for 7 levels)
    coeff_kernel<<<(BATCH * NFRAMES + 255) / 256, 256, 0, stream>>>(nin, nzf_w, nzf_b, cof);

    // ---- harmonic + filtered noise output ----
    harm_noise_kernel<<<BATCH * (BAND / 256), 256, 0, stream>>>(Fb, Ab, Bb, cof, noise, outp);
}